// WaveCell_12317966205374
// MI455X (gfx1250) — compile-verified
//
#include <hip/hip_runtime.h>

// CDNA5 / gfx1250, wave32.
typedef float v2f __attribute__((ext_vector_type(2)));
typedef float v8f __attribute__((ext_vector_type(8)));

#define GN 512      // grid is GN x GN
#define LDSP 17     // padded LDS row stride: conflict-free transposed reads

// Precompute time-invariant PML coefficients:
//   C1 = 1/(inv_dt2 + b*(0.5/DT)) = 1/(4+b),  C3 = inv_dt2 - b*(0.5/DT) = 4-b
__global__ __launch_bounds__(256)
void pml_coeff(const float* __restrict__ b,
               float* __restrict__ C1p, float* __restrict__ C3p)
{
  const int i = blockIdx.x * 256 + threadIdx.x;    // grid = 1024 blocks
  const float bv = b[i];
  C1p[i] = 1.0f / (4.0f + bv);
  C3p[i] = 4.0f - bv;
}

// One explicit time step:
//   un = C1 * (C2*u1 - C3*u2 + c2*Lap(u1) + mask*xi)
// Lap per 16x16 tile = T*U + U*T on V_WMMA_F32_16X16X4_F32.
// T = tridiag(1,-2,1) is symmetric, so ONE v2f register pair is both the
// A-operand chunk of T*U (T[lp,kA]) and the B-operand chunk of U*T (T[kA,lp]).
// T row entries {0,1,-2} are encoded as 2-bit two's-complement fields in a
// single 32-bit code per lane: decode = shl / ashr(30) / cvt_f32_i32.
__global__ __launch_bounds__(256)
void wave_step(const float* __restrict__ xi,
               const float* __restrict__ xi_next,
               const float* __restrict__ u1,
               const float* __restrict__ u2,
               const float* __restrict__ c2,
               const float* __restrict__ msk,
               const float* __restrict__ C1p,
               const float* __restrict__ C3p,
               float* __restrict__ out)
{
  __shared__ float lds[8][16 * LDSP];              // 8.5 KB / block, per-wave slices

  const int lane = threadIdx.x & 31;
  const int wave = threadIdx.x >> 5;
  const int tile = blockIdx.x * 8 + wave;          // 0..1023, one 16x16 tile per wave
  const int r0 = (tile >> 5) * 16;
  const int c0 = (tile & 31) * 16;

  const int hi = lane >> 4;                        // lanes 16-31: K+2 / M+8 halves
  const int lp = lane & 15;

  float* t = lds[wave];  // private to this wave -> no block barrier required

  // T row of this lane, 2-bit signed fields: entry k at bits [2k+1:2k].
  // Pattern (1,-2,1) = 01,10,01 centered at k=lp; shifts clip the k=-1/k=16 taps.
  const unsigned rowcode = (unsigned)((0x19ull << (2 * lp)) >> 2);

  // ---- stage u1 tile into LDS, fully coalesced (2 x b128 per lane) -------
  {
    const int lr = lane >> 2;                      // 0..7
    const int lc = (lane & 3) * 4;                 // 0,4,8,12 (16B aligned)
    const float4 p0 = *(const float4*)&u1[(r0 + lr) * GN + c0 + lc];
    const float4 p1 = *(const float4*)&u1[(r0 + 8 + lr) * GN + c0 + lc];
    float* q0 = &t[lr * LDSP + lc];
    q0[0] = p0.x; q0[1] = p0.y; q0[2] = p0.z; q0[3] = p0.w;
    float* q1 = &t[(lr + 8) * LDSP + lc];
    q1[0] = p1.x; q1[1] = p1.y; q1[2] = p1.z; q1[3] = p1.w;
  }
  // Same-wave LDS ops are in-order; compiler inserts s_wait_dscnt before reads.

  // ---- Laplacian via block matmuls: acc = T*U + U*T -----------------------
  v8f acc = {};
  #pragma unroll
  for (int k0 = 0; k0 < 16; k0 += 4) {
    const int sh = 30 - 2 * k0 - 4 * hi;           // field {kA, kA+1}, kA = k0+2*hi

    v2f tv;                                        // T chunk: A of T*U == B of U*T
    tv.x = (float)((int)(rowcode << sh) >> 30);
    tv.y = (float)((int)(rowcode << (sh - 2)) >> 30);

    // T * U : B = U[k0:k0+4, :] (LDS rows)
    v2f bU;
    bU.x = t[(k0 + 2 * hi + 0) * LDSP + lp];
    bU.y = t[(k0 + 2 * hi + 1) * LDSP + lp];
    acc = __builtin_amdgcn_wmma_f32_16x16x4_f32(
        false, tv, false, bU, (short)0, acc, false, false);

    // U * T : A = U[:, k0:k0+4] (LDS columns, pad => conflict-free)
    v2f aU;
    aU.x = t[lp * LDSP + k0 + 2 * hi + 0];
    aU.y = t[lp * LDSP + k0 + 2 * hi + 1];
    acc = __builtin_amdgcn_wmma_f32_16x16x4_f32(
        false, aU, false, tv, (short)0, acc, false, false);
  }

  // ---- halo couplings + PML pointwise update + next-slice prefetch --------
  #pragma unroll
  for (int v = 0; v < 8; ++v) {
    const int M  = v + 8 * hi;                     // C/D layout: VGPR v, lane half
    const int gr = r0 + M;
    const int gc = c0 + lp;
    const int idx = gr * GN + gc;

    float lap = acc[v];
    if (M == 0  && gr > 0)      lap += u1[(gr - 1) * GN + gc];
    if (M == 15 && gr < GN - 1) lap += u1[(gr + 1) * GN + gc];
    if (lp == 0  && gc > 0)      lap += u1[idx - 1];
    if (lp == 15 && gc < GN - 1) lap += u1[idx + 1];

    const float u1c = t[M * LDSP + lp];            // tile center from LDS
    out[idx] = C1p[idx] * (8.0f * u1c - C3p[idx] * u2[idx]
                           + c2[idx] * lap + msk[idx] * xi[idx]);

    // Stream-ahead on the only true HBM traffic: next forcing slice
    // (one 64B line per tile row).
    if (lp == 0) __builtin_prefetch(&xi_next[gr * GN + c0], 0, 1);
  }
}

extern "C" void kernel_launch(void* const* d_in, const int* in_sizes, int n_in,
                              void* d_out, int out_size, void* d_ws, size_t ws_size,
                              hipStream_t stream) {
  const float* inp = (const float*)d_in[0];        // (NT,1,512,512)
  const float* un1 = (const float*)d_in[1];        // (1,1,512,512) initial u1
  const float* un2 = (const float*)d_in[2];        // (1,1,512,512) initial u2
  const float* c2  = (const float*)d_in[3];        // (512,512)
  const float* msk = (const float*)d_in[4];        // (512,512)
  const float* b   = (const float*)d_in[5];        // (1,1,512,512)

  const int plane = GN * GN;
  const int nt = in_sizes[0] / plane;              // 256

  float* buf0 = (float*)d_ws;                      // ping
  float* buf1 = buf0 + plane;                      // pong
  float* C1p  = buf1 + plane;                      // PML coeff planes (4 MB ws total)
  float* C3p  = C1p  + plane;
  float* outp = (float*)d_out;

  // Hoist time-invariant PML coefficients (removes 8 f32 divides per wave
  // from each of the 256 hot steps). Deterministic; graph-capture safe.
  pml_coeff<<<dim3(plane / 256), dim3(256), 0, stream>>>(b, C1p, C3p);

  dim3 grid(128);                                  // 128 blocks * 8 waves = 1024 tiles
  dim3 block(256);

  for (int t = 0; t < nt; ++t) {
    const float* xi      = inp + (size_t)t * plane;
    const float* xi_next = (t + 1 < nt) ? (inp + (size_t)(t + 1) * plane) : xi;

    // carry: u1 = out(t-1), u2 = out(t-2); out(t) = buf[t%2] (d_out for last).
    const float* u1 = (t == 0) ? un1 : ((t & 1) ? buf0 : buf1);
    const float* u2 = (t == 0) ? un2 : ((t == 1) ? un1 : ((t & 1) ? buf1 : buf0));
    float* o = (t == nt - 1) ? outp : ((t & 1) ? buf1 : buf0);

    // For t>=2 the output buffer aliases u2; u2 is only read pointwise at the
    // writing cell, so the in-place update is race-free.
    wave_step<<<grid, block, 0, stream>>>(xi, xi_next, u1, u2, c2, msk,
                                          C1p, C3p, o);
  }
}